// TripletLoss_20804821582533
// MI455X (gfx1250) — compile-verified
//
#include <hip/hip_runtime.h>

typedef __attribute__((ext_vector_type(2))) float v2f;
typedef __attribute__((ext_vector_type(8))) float v8f;

#define N_ROWS 96
#define DIM 768
#define NN (N_ROWS * N_ROWS)   // 9216
#define P_COUNT 9120.0         // 2 * 96*95/2
#define THRESH 0.3f

// ---------------------------------------------------------------------------
// Kernel 1: L2-normalize rows of both inputs (torch eps clamp on the norm).
// ---------------------------------------------------------------------------
__global__ __launch_bounds__(256) void normalize_rows(
    const float* __restrict__ stereos, const float* __restrict__ astereos,
    float* __restrict__ xn, float* __restrict__ yn) {
  int row = blockIdx.x;  // 0..191
  const float* src = (row < N_ROWS) ? (stereos + row * DIM)
                                    : (astereos + (row - N_ROWS) * DIM);
  float* dst = (row < N_ROWS) ? (xn + row * DIM) : (yn + (row - N_ROWS) * DIM);

  __shared__ float red[256];
  float ss = 0.0f;
  for (int i = threadIdx.x; i < DIM; i += 256) {
    float v = src[i];
    ss += v * v;
  }
  red[threadIdx.x] = ss;
  __syncthreads();
  for (int off = 128; off > 0; off >>= 1) {
    if (threadIdx.x < off) red[threadIdx.x] += red[threadIdx.x + off];
    __syncthreads();
  }
  float denom = fmaxf(sqrtf(red[0]), 1e-8f);
  for (int i = threadIdx.x; i < DIM; i += 256) dst[i] = src[i] / denom;
}

// ---------------------------------------------------------------------------
// Kernel 2: three 96x96 Gram matrices via V_WMMA_F32_16X16X4_F32.
// One wave per 16x16 tile: blocks = 3 matrices * 36 tiles = 108, 32 thr each.
//   mat 0: Xn @ Xn^T   (stereos self)
//   mat 1: Yn @ Yn^T   (astereos self)
//   mat 2: Yn @ Xn^T   (cross / negatives)
// A (16x4 f32) lane layout: lane L holds row M=L&15; VGPR0={K0,K2}, VGPR1={K1,K3}
// split by lane half. B (4x16) is symmetric with N=L&15 since B cols are rows
// of a row-major matrix. C/D: 8 VGPRs, M = v + 8*(L>>4), N = L&15.
// ---------------------------------------------------------------------------
__global__ __launch_bounds__(32) void gram_wmma(
    const float* __restrict__ xn, const float* __restrict__ yn,
    float* __restrict__ sims) {
  int b = blockIdx.x;       // 0..107
  int mat = b / 36;
  int t = b % 36;
  int tm = t / 6;
  int tn = t % 6;

  const float* Arows = (mat == 0) ? xn : yn;
  const float* Brows = (mat == 1) ? yn : xn;

  int lane = threadIdx.x;
  int li = lane & 15;
  int kh = (lane >> 4) << 1;  // 0 for lanes 0-15, 2 for lanes 16-31

  const float* arow = Arows + (tm * 16 + li) * DIM + kh;
  const float* brow = Brows + (tn * 16 + li) * DIM + kh;

  v8f c = {};
  for (int k = 0; k < DIM; k += 4) {
    v2f av = *(const v2f*)(arow + k);
    v2f bv = *(const v2f*)(brow + k);
    c = __builtin_amdgcn_wmma_f32_16x16x4_f32(
        /*neg_a=*/false, av, /*neg_b=*/false, bv,
        /*c_mod=*/(short)0, c, /*reuse_a=*/false, /*reuse_b=*/false);
  }

  float* out = sims + mat * NN + (tm * 16 + 8 * (lane >> 4)) * N_ROWS
             + tn * 16 + li;
#pragma unroll
  for (int v = 0; v < 8; ++v) out[v * N_ROWS] = c[v];
}

// ---------------------------------------------------------------------------
// Kernel 3: hinge partial sums. One block per (pos-matrix, row) = 192 blocks.
// Negatives (9216 f32 = 36 KB) staged in LDS; each active thread owns one
// upper-triangle column of its row and streams all 9216 negatives.
// ---------------------------------------------------------------------------
__global__ __launch_bounds__(256) void hinge_partial(
    const float* __restrict__ sims, float* __restrict__ partials) {
  __shared__ float neg[NN];
  __shared__ double red[256];

  const float* negsrc = sims + 2 * NN;
  for (int i = threadIdx.x; i < NN; i += 256) neg[i] = negsrc[i];
  __syncthreads();

  int mat = blockIdx.x / N_ROWS;  // 0 = stereos-self, 1 = astereos-self
  int row = blockIdx.x % N_ROWS;
  const float* prow = sims + mat * NN + row * N_ROWS;

  double acc = 0.0;
  for (int col = row + 1 + (int)threadIdx.x; col < N_ROWS; col += 256) {
    float t = THRESH - prow[col];
    float s = 0.0f;
#pragma unroll 8
    for (int j = 0; j < NN; ++j) s += fmaxf(0.0f, neg[j] + t);
    acc += (double)s;
  }

  red[threadIdx.x] = acc;
  __syncthreads();
  for (int off = 128; off > 0; off >>= 1) {
    if (threadIdx.x < off) red[threadIdx.x] += red[threadIdx.x + off];
    __syncthreads();
  }
  if (threadIdx.x == 0) partials[blockIdx.x] = (float)red[0];
}

// ---------------------------------------------------------------------------
// Kernel 4: deterministic ordered reduction of the 192 partials -> mean.
// ---------------------------------------------------------------------------
__global__ void finalize(const float* __restrict__ partials,
                         float* __restrict__ out) {
  if (threadIdx.x == 0 && blockIdx.x == 0) {
    double s = 0.0;
    for (int i = 0; i < 192; ++i) s += (double)partials[i];
    out[0] = (float)(s / (P_COUNT * (double)NN));
  }
}

// ---------------------------------------------------------------------------
extern "C" void kernel_launch(void* const* d_in, const int* in_sizes, int n_in,
                              void* d_out, int out_size, void* d_ws,
                              size_t ws_size, hipStream_t stream) {
  const float* stereos  = (const float*)d_in[0];  // [96,768]
  const float* astereos = (const float*)d_in[1];  // [96,768]
  float* out = (float*)d_out;

  float* ws = (float*)d_ws;
  float* xn       = ws;                       // 96*768
  float* yn       = ws + N_ROWS * DIM;        // 96*768
  float* sims     = ws + 2 * N_ROWS * DIM;    // 3*9216
  float* partials = sims + 3 * NN;            // 192

  normalize_rows<<<192, 256, 0, stream>>>(stereos, astereos, xn, yn);
  gram_wmma<<<108, 32, 0, stream>>>(xn, yn, sims);
  hinge_partial<<<192, 256, 0, stream>>>(sims, partials);
  finalize<<<1, 32, 0, stream>>>(partials, out);
}